// BModel_15161234555476
// MI455X (gfx1250) — compile-verified
//
#include <hip/hip_runtime.h>

// Sliding 100x100 mean over [32,1,1124,1124] fp32 -> [32,1,1025,1025].
//
// Roofline: ~162 MB in + ~134 MB out => ~12.7 us @ 23.3 TB/s. The kernel must
// also stay instruction-lean (~<30M wave-inst). So: two passes of O(1)
// running-window recurrences (no per-row scans, no per-row block barriers in
// pass 1), with the 148 MB horizontal-sum intermediate living in the 192 MB L2.
//
// Pass 1 (horizontal): thread per (row, quarter-row segment); serial walk with
//   s(c+1) = s(c) + x[c+100] - x[c], vectorized x4 with b128 NT loads.
// Pass 2 (vertical): block owns 256 columns x ~257 output rows; each R row is
//   DMA'd to a depth-102 LDS ring with GLOBAL_LOAD_ASYNC_TO_LDS_B128
//   (prefetch depth 2, s_wait_asynccnt), per-thread running column sums,
//   nontemporal output stores. R is read exactly once.
//
// Fallback: if ws_size can't hold R, run the fused single-kernel version.

namespace {
constexpr int KW      = 100;             // window
constexpr int IH      = 1124;
constexpr int IW      = 1124;
constexpr int OH      = IH - KW + 1;     // 1025
constexpr int OW      = IW - KW + 1;     // 1025
constexpr int RSTRIDE = 1028;            // padded R row (16B-aligned rows)
constexpr int SEGW    = 260;             // pass-1 cols per segment (4 segs)
constexpr int NSEG1   = 4;
constexpr int TW2     = 256;             // pass-2 columns per block
constexpr int NT2     = 128;             // pass-2 threads (4 waves, 2 cols/thr)
constexpr int RD      = 102;             // LDS ring depth (100 + prefetch 2)
constexpr int YSEG    = 257;             // pass-2 output rows per block (4 segs)
constexpr float INVK2 = 1.0f / (float)(KW * KW);
}

typedef float f4 __attribute__((ext_vector_type(4)));
typedef int v4i __attribute__((vector_size(16)));
typedef __attribute__((address_space(1))) v4i* gv4i_p;  // global
typedef __attribute__((address_space(3))) v4i* lv4i_p;  // LDS

#if __has_builtin(__builtin_amdgcn_global_load_async_to_lds_b128)
#define USE_ASYNC_LDS 1
#else
#define USE_ASYNC_LDS 0
#endif

#if __has_builtin(__builtin_amdgcn_s_wait_asynccnt)
#define WAIT_ASYNCCNT(n) __builtin_amdgcn_s_wait_asynccnt(n)
#else
#define WAIT_ASYNCCNT(n) asm volatile("s_wait_asynccnt %0" :: "i"(n) : "memory")
#endif

// ---------------------------------------------------------------------------
// Pass 1: horizontal 100-window sums. R[b, y, c] = sum_{dx<100} x[b, y, c+dx]
// ---------------------------------------------------------------------------
__global__ __launch_bounds__(256) void hpass_kernel(
    const float* __restrict__ x, float* __restrict__ r, int nTask)
{
  const int tid = blockIdx.x * 256 + threadIdx.x;
  if (tid >= nTask) return;
  const int seg = tid & (NSEG1 - 1);
  const int row = (tid >> 2) % IH;
  const int b   = tid / (IH * NSEG1);

  const float* __restrict__ xr = x + ((size_t)b * IH + row) * IW;   // 16B-aligned
  float* __restrict__ rr       = r + ((size_t)b * IH + row) * RSTRIDE;

  const int cs = seg * SEGW;                 // multiple of 4
  const int ce = min(OW, cs + SEGW);         // 260,260,260,245

  // init: window sum over x[cs .. cs+99] (25 aligned NT b128 loads)
  float s = 0.0f;
#pragma unroll
  for (int k = 0; k < KW; k += 4) {
    const f4 q = __builtin_nontemporal_load((const f4*)(xr + cs + k));
    s += q.x + q.y + q.z + q.w;
  }

  // main: 4 outputs per iteration, aligned b128 traffic
  int c = cs;
  while (c + 3 < ce) {
    const f4 xa = __builtin_nontemporal_load((const f4*)(xr + c));
    const f4 xb = __builtin_nontemporal_load((const f4*)(xr + c + KW));
    f4 o;
    o.x = s;
    o.y = o.x + xb.x - xa.x;
    o.z = o.y + xb.y - xa.y;
    o.w = o.z + xb.z - xa.z;
    *(f4*)(rr + c) = o;                      // R stays in L2 (default RT)
    s = o.w + xb.w - xa.w;
    c += 4;
  }
  while (c < ce) {                           // <=1 scalar tail (seg 3)
    rr[c] = s;
    if (c + KW < IW) s += xr[c + KW] - xr[c];
    ++c;
  }
}

// ---------------------------------------------------------------------------
// Pass 2: vertical 100-window mean over R, async-DMA'd LDS ring buffer.
// ---------------------------------------------------------------------------
__global__ __launch_bounds__(NT2) void vpass_kernel(
    const float* __restrict__ r, float* __restrict__ out)
{
  const int ct = blockIdx.x;              // 0..4 column tiles
  const int ys = blockIdx.y;              // 0..3 row segments
  const int b  = blockIdx.z;

  const int c0 = ct * TW2;
  const int y0 = ys * YSEG;
  const int nS = min(YSEG, OH - y0);
  const int t  = threadIdx.x;

  __shared__ __align__(16) float ring[RD][TW2];   // 104448 B

  const float* __restrict__ rb = r + (size_t)b * IH * RSTRIDE;
  float* __restrict__ ob       = out + (size_t)b * OH * OW;

  const int yiEnd = y0 + nS + KW - 2;     // last R row consumed (inclusive)

  auto stage = [&](int yi) {
#if USE_ASYNC_LDS
    if (t < TW2 / 4) {
      const int cc = c0 + t * 4;
      if (cc + 3 < RSTRIDE) {             // stay inside padded R row
        __builtin_amdgcn_global_load_async_to_lds_b128(
            (gv4i_p)(rb + (size_t)yi * RSTRIDE + cc),
            (lv4i_p)(&ring[yi % RD][t * 4]),
            0, 0);
      }
    }
#else
    for (int i = t; i < TW2; i += NT2) {
      const int cc = c0 + i;
      ring[yi % RD][i] = (cc < RSTRIDE) ? rb[(size_t)yi * RSTRIDE + cc] : 0.0f;
    }
#endif
  };

  // prime the pipeline: rows y0, y0+1 in flight
  stage(y0);
  stage(y0 + 1);

  float v0 = 0.0f, v1 = 0.0f;             // running vertical sums (2 cols/thr)
  const int ca = c0 + t, cb = c0 + t + NT2;
  const bool oka = ca < OW, okb = cb < OW;

  for (int yi = y0; yi <= yiEnd; ++yi) {
    // keep 2 rows in flight; wait until row yi has landed (in-order completion)
    if (yi + 2 <= yiEnd) { stage(yi + 2); WAIT_ASYNCCNT(2); }
    else if (yi + 1 <= yiEnd) { WAIT_ASYNCCNT(1); }
    else { WAIT_ASYNCCNT(0); }
    __syncthreads();                      // row yi visible to all waves

    const int sNew = yi % RD;
    v0 += ring[sNew][t];
    v1 += ring[sNew][t + NT2];

    if (yi >= y0 + KW - 1) {
      const int yo = yi - (KW - 1);
      if (oka) __builtin_nontemporal_store(v0 * INVK2, &ob[(size_t)yo * OW + ca]);
      if (okb) __builtin_nontemporal_store(v1 * INVK2, &ob[(size_t)yo * OW + cb]);
      const int sOld = yo % RD;           // row leaving the window
      v0 -= ring[sOld][t];
      v1 -= ring[sOld][t + NT2];
    }
    __syncthreads();                      // consumption done before slot reuse
  }
}

// ---------------------------------------------------------------------------
// Fallback: fused single-kernel version (used only if ws can't hold R).
// ---------------------------------------------------------------------------
namespace {
constexpr int FTW   = 128;
constexpr int FNSEG = FTW + KW;           // 228
constexpr int FNTHR = 256;
constexpr int FRSEG = 352;
}

__global__ __launch_bounds__(FNTHR) void fused_kernel(
    const float* __restrict__ x, float* __restrict__ out)
{
  const int x0 = blockIdx.x * FTW;
  const int y0 = blockIdx.y * FRSEG;
  const int b  = blockIdx.z;
  const int nR   = min(FRSEG, OH - y0);
  const int nOut = min(FTW, OW - x0);
  const int t = threadIdx.x, lane = t & 31, wave = t >> 5;

  __shared__ float ring[KW][FTW];
  __shared__ __align__(16) float seg[2][FNSEG + 4];
  __shared__ float pbuf[FNSEG];
  __shared__ float wsum[FNTHR / 32];

  const float* __restrict__ xb = x + (size_t)b * IH * IW;
  float* __restrict__ ob = out + (size_t)b * OH * OW;
  const int yiEnd = y0 + nR + KW - 2;
  float vsum = 0.0f;

  auto stage = [&](int yi, int buf) {
#if USE_ASYNC_LDS
    if (t < FNSEG / 4) {
      const int xcol = x0 + t * 4;
      if (xcol + 3 < IW) {
        __builtin_amdgcn_global_load_async_to_lds_b128(
            (gv4i_p)(xb + (size_t)yi * IW + xcol),
            (lv4i_p)(&seg[buf][t * 4]), 0, 0);
      }
    }
#else
    for (int i = t; i < FNSEG; i += FNTHR) {
      const int xcol = x0 + i;
      seg[buf][i] = (xcol < IW) ? xb[(size_t)yi * IW + xcol] : 0.0f;
    }
#endif
  };

  stage(y0, y0 & 1);
  WAIT_ASYNCCNT(0);
  __syncthreads();

  for (int yi = y0; yi <= yiEnd; ++yi) {
    const int cur = yi & 1;
    if (yi < yiEnd) stage(yi + 1, (yi + 1) & 1);

    float v = (t < FNSEG) ? seg[cur][t] : 0.0f;
#pragma unroll
    for (int d = 1; d < 32; d <<= 1) {
      const float n = __shfl_up(v, d, 32);
      if (lane >= d) v += n;
    }
    if (lane == 31) wsum[wave] = v;
    __syncthreads();
    float off = 0.0f;
#pragma unroll
    for (int w2 = 0; w2 < FNTHR / 32; ++w2) off += (w2 < wave) ? wsum[w2] : 0.0f;
    v += off;
    if (t < FNSEG) pbuf[t] = v;
    __syncthreads();

    if (t < nOut) {
      const float rs = pbuf[t + KW - 1] - ((t > 0) ? pbuf[t - 1] : 0.0f);
      ring[yi % KW][t] = rs;
      vsum += rs;
      if (yi >= y0 + KW - 1) {
        const int yo = yi - (KW - 1);
        ob[(size_t)yo * OW + (x0 + t)] = vsum * INVK2;
        vsum -= ring[yo % KW][t];
      }
    }
    WAIT_ASYNCCNT(0);
    __syncthreads();
  }
}

extern "C" void kernel_launch(void* const* d_in, const int* in_sizes, int n_in,
                              void* d_out, int out_size, void* d_ws, size_t ws_size,
                              hipStream_t stream) {
  (void)n_in; (void)out_size;
  const float* x = (const float*)d_in[0];
  float* out = (float*)d_out;
  const int B = in_sizes[0] / (IH * IW);                 // 32

  const size_t rBytes = (size_t)B * IH * RSTRIDE * sizeof(float);  // ~148 MB
  if (d_ws != nullptr && ws_size >= rBytes) {
    float* r = (float*)d_ws;
    const int nTask = B * IH * NSEG1;
    hpass_kernel<<<dim3((nTask + 255) / 256), 256, 0, stream>>>(x, r, nTask);
    vpass_kernel<<<dim3((OW + TW2 - 1) / TW2,            // 5 column tiles
                       (OH + YSEG - 1) / YSEG,           // 4 row segments
                       B),
                  NT2, 0, stream>>>(r, out);
  } else {
    fused_kernel<<<dim3((OW + FTW - 1) / FTW,
                        (OH + FRSEG - 1) / FRSEG,
                        B),
                   FNTHR, 0, stream>>>(x, out);
  }
}